// SimPoolAggregator_31439160607285
// MI455X (gfx1250) — compile-verified
//
#include <hip/hip_runtime.h>

// ---------------------------------------------------------------------------
// SimPool aggregator for MI455X (gfx1250, wave32).
// Memory-bound after algebraic collapse of the kk GEMM; WMMA fp32 16x16x4
// used for the three small dense GEMMs (qq, q2, sims).
// ---------------------------------------------------------------------------

#define DIMD 512
#define BATCH 32
#define GAMMA_C 1.25f
#define INV_GAMMA_C 0.8f
#define EPS_C 1e-6f
#define LN_EPS_C 1e-6f

typedef float v2f __attribute__((ext_vector_type(2)));
typedef float v8f __attribute__((ext_vector_type(8)));

__device__ __forceinline__ float wave_sum(float v) {
#pragma unroll
  for (int o = 16; o > 0; o >>= 1) v += __shfl_xor(v, o, 32);
  return v;
}
__device__ __forceinline__ float wave_min(float v) {
#pragma unroll
  for (int o = 16; o > 0; o >>= 1) v = fminf(v, __shfl_xor(v, o, 32));
  return v;
}
// monotone float <-> uint map so integer atomicMin orders like float min
__device__ __forceinline__ unsigned enc_f(float f) {
  unsigned u = __float_as_uint(f);
  unsigned mask = (unsigned)(-(int)(u >> 31)) | 0x80000000u;
  return u ^ mask;
}
__device__ __forceinline__ float dec_f(unsigned e) {
  unsigned mask = ((e >> 31) - 1u) | 0x80000000u;
  return __uint_as_float(e ^ mask);
}

// ---------------------------------------------------------------------------
__global__ void k_init(float* zero_region, unsigned* vmins) {
  int i = blockIdx.x * 256 + threadIdx.x;
  if (i < 4 * BATCH * DIMD) zero_region[i] = 0.0f;
  if (i < 2) vmins[i] = 0xFFFFFFFFu;
}

// ---------------------------------------------------------------------------
// Pass 1: one stream over x. Per patch n: mean/rsigma of LayerNorm.
// Cross-lane reduce per d iteration -> GAP[b,d] partials (per-wave LDS slice).
__global__ void k_gap_stats(const float* __restrict__ x, float* __restrict__ gap,
                            float* __restrict__ mu_o, float* __restrict__ rs_o,
                            int N) {
  __shared__ float sAcc[8][DIMD];
  int tid = threadIdx.x;
  int b = blockIdx.x;
  int n = blockIdx.y * 256 + tid;
  int w = tid >> 5, lane = tid & 31;
  const float* xb = x + ((size_t)b * DIMD) * N + n;
  float s1 = 0.0f, s2 = 0.0f;
  for (int d = 0; d < DIMD; ++d) {
    float v = xb[(size_t)d * N];
    s1 += v;
    s2 += v * v;
    float ws = wave_sum(v);
    if (lane == 0) sAcc[w][d] = ws;
  }
  __syncthreads();
  float invN = 1.0f / (float)N;
  for (int i = tid; i < DIMD; i += 256) {
    float t = 0.0f;
#pragma unroll
    for (int j = 0; j < 8; ++j) t += sAcc[j][i];
    atomicAdd(&gap[b * DIMD + i], t * invN);
  }
  float mu = s1 * (1.0f / DIMD);
  float var = s2 * (1.0f / DIMD) - mu * mu;
  size_t idx = (size_t)b * N + n;
  mu_o[idx] = mu;
  rs_o[idx] = rsqrtf(var + LN_EPS_C);
}

// ---------------------------------------------------------------------------
// fp32 WMMA GEMM: C[32 x Ntot] = A[32 x 512] * op(W), one wave per 16x16 tile.
// TRANSW=true : B[k][n] = W[n][k] (row-major weight, contract over its columns)
// TRANSW=false: B[k][n] = W[k][n]
// A 16x4 fp32 layout: lane L -> m = L&15, kbase = 2*(L>>4); a[v] = A[m][k+kbase+v]
// B 4x16 fp32 layout mirrored; C/D: VGPR r -> row r + 8*(L>>4), col L&15.
// Compile-time TRANSW keeps the K-loop branch-free; bases hoisted so the body
// is constant-offset b64 loads feeding v_wmma.
template <bool TRANSW>
__global__ void k_gemm32(const float* __restrict__ A, const float* __restrict__ W,
                         float* __restrict__ C, int ldC, float scale) {
  int lane = threadIdx.x;
  int m0 = blockIdx.y * 16, n0 = blockIdx.x * 16;
  int r = lane & 15;
  int kh2 = (lane >> 4) << 1;  // 0 or 2
  const float* aRow = A + (m0 + r) * DIMD + kh2;  // contiguous pair at +k
  const float* wBase = TRANSW ? (W + (n0 + r) * DIMD + kh2)  // contiguous pair
                              : (W + n0 + r + (size_t)kh2 * DIMD);
  v8f acc = {0.f, 0.f, 0.f, 0.f, 0.f, 0.f, 0.f, 0.f};
#pragma unroll 8
  for (int k = 0; k < DIMD; k += 4) {
    v2f a = *(const v2f*)(aRow + k);
    v2f bb;
    if (TRANSW) {
      bb = *(const v2f*)(wBase + k);
    } else {
      bb.x = wBase[(size_t)k * DIMD];
      bb.y = wBase[(size_t)(k + 1) * DIMD];
    }
    acc = __builtin_amdgcn_wmma_f32_16x16x4_f32(false, a, false, bb,
                                                (short)0, acc, false, false);
  }
#pragma unroll
  for (int rr = 0; rr < 8; ++rr) {
    C[(m0 + rr + ((lane >> 4) << 3)) * ldC + n0 + r] = acc[rr] * scale;
  }
}

// ---------------------------------------------------------------------------
// Pass 3: second stream over x. logit[b,n] = kv . q2[b], running global min(kv).
__global__ void k_logits(const float* __restrict__ x, const float* __restrict__ mu_a,
                         const float* __restrict__ rs_a, const float* __restrict__ q2,
                         const float* __restrict__ lnw, const float* __restrict__ lnb,
                         float* __restrict__ logits, unsigned* __restrict__ vmin,
                         int N) {
  __shared__ float sW[DIMD], sB[DIMD], sQ[DIMD];
  __shared__ float sMin[8];
  int tid = threadIdx.x, b = blockIdx.x;
  int n = blockIdx.y * 256 + tid;
  for (int i = tid; i < DIMD; i += 256) {
    sW[i] = lnw[i];
    sB[i] = lnb[i];
    sQ[i] = q2[b * DIMD + i];
  }
  __syncthreads();
  size_t idx = (size_t)b * N + n;
  float mu = mu_a[idx], rs = rs_a[idx];
  const float* xb = x + ((size_t)b * DIMD) * N + n;
  float acc = 0.0f, mn = __builtin_inff();
  for (int d = 0; d < DIMD; ++d) {
    float v = xb[(size_t)d * N];
    float kvv = (v - mu) * rs * sW[d] + sB[d];
    acc = fmaf(kvv, sQ[d], acc);
    mn = fminf(mn, kvv);
  }
  logits[idx] = acc;
  float wmn = wave_min(mn);
  if ((tid & 31) == 0) sMin[tid >> 5] = wmn;
  __syncthreads();
  if (tid == 0) {
    float m = sMin[0];
#pragma unroll
    for (int j = 1; j < 8; ++j) m = fminf(m, sMin[j]);
    atomicMin(vmin, enc_f(m));
  }
}

// ---------------------------------------------------------------------------
// Pass 4: row softmax over N (in place), native v_exp_f32.
__global__ void k_softmax(float* __restrict__ logits, int N) {
  __shared__ float red[256];
  int tid = threadIdx.x, b = blockIdx.x;
  float* row = logits + (size_t)b * N;
  float m = -__builtin_inff();
  for (int i = tid; i < N; i += 256) m = fmaxf(m, row[i]);
  red[tid] = m;
  __syncthreads();
  for (int s = 128; s > 0; s >>= 1) {
    if (tid < s) red[tid] = fmaxf(red[tid], red[tid + s]);
    __syncthreads();
  }
  m = red[0];
  __syncthreads();
  float s = 0.0f;
  for (int i = tid; i < N; i += 256) {
    float e = __builtin_amdgcn_exp2f((row[i] - m) * 1.4426950408889634f);
    row[i] = e;
    s += e;
  }
  red[tid] = s;
  __syncthreads();
  for (int st = 128; st > 0; st >>= 1) {
    if (tid < st) red[tid] += red[tid + st];
    __syncthreads();
  }
  float inv = 1.0f / red[0];
  for (int i = tid; i < N; i += 256) row[i] *= inv;
}

// ---------------------------------------------------------------------------
// Pass 5: third stream over x. pooled[b,d] += attn[b,n]*(kv - vmin + eps)^1.25
__global__ void k_pooled(const float* __restrict__ x, const float* __restrict__ mu_a,
                         const float* __restrict__ rs_a, const float* __restrict__ attn,
                         const float* __restrict__ lnw, const float* __restrict__ lnb,
                         const unsigned* __restrict__ vminp,
                         float* __restrict__ pooled, int N) {
  __shared__ float sW[DIMD], sB[DIMD];
  __shared__ float sAcc[8][DIMD];
  int tid = threadIdx.x, b = blockIdx.x;
  int n = blockIdx.y * 256 + tid;
  int w = tid >> 5, lane = tid & 31;
  for (int i = tid; i < DIMD; i += 256) {
    sW[i] = lnw[i];
    sB[i] = lnb[i];
  }
  __syncthreads();
  float vmin = dec_f(*vminp);
  size_t idx = (size_t)b * N + n;
  float mu = mu_a[idx], rs = rs_a[idx], at = attn[idx];
  const float* xb = x + ((size_t)b * DIMD) * N + n;
  for (int d = 0; d < DIMD; ++d) {
    float v = xb[(size_t)d * N];
    float kvv = (v - mu) * rs * sW[d] + sB[d];
    float sft = kvv - vmin + EPS_C;  // strictly positive
    float p = __builtin_amdgcn_exp2f(GAMMA_C * __builtin_amdgcn_logf(sft));
    float ws = wave_sum(at * p);
    if (lane == 0) sAcc[w][d] = ws;
  }
  __syncthreads();
  for (int i = tid; i < DIMD; i += 256) {
    float t = 0.0f;
#pragma unroll
    for (int j = 0; j < 8; ++j) t += sAcc[j][i];
    atomicAdd(&pooled[b * DIMD + i], t);
  }
}

// ---------------------------------------------------------------------------
__global__ void k_finalize(float* __restrict__ pooled, int count) {
  int i = blockIdx.x * 256 + threadIdx.x;
  if (i < count) {
    float p = pooled[i];
    pooled[i] = __builtin_amdgcn_exp2f(INV_GAMMA_C * __builtin_amdgcn_logf(p));
  }
}

// ---------------------------------------------------------------------------
extern "C" void kernel_launch(void* const* d_in, const int* in_sizes, int n_in,
                              void* d_out, int out_size, void* d_ws, size_t ws_size,
                              hipStream_t stream) {
  (void)in_sizes; (void)n_in; (void)out_size; (void)ws_size;
  const float* audio = (const float*)d_in[0];  // (32,512,64,96)
  const float* image = (const float*)d_in[1];  // (32,512,32,32)
  const float* lnw = (const float*)d_in[2];
  const float* lnb = (const float*)d_in[3];
  const float* wq = (const float*)d_in[4];
  const float* wk = (const float*)d_in[5];
  float* out = (float*)d_out;  // (32,32)

  const int NA = 64 * 96;  // 6144
  const int NI = 32 * 32;  // 1024
  const int BD = BATCH * DIMD;

  float* ws = (float*)d_ws;
  float* gap_a = ws;               // [BD]   (zeroed)
  float* gap_i = ws + 1 * BD;      // [BD]   (zeroed)
  float* pool_a = ws + 2 * BD;     // [BD]   (zeroed)
  float* pool_i = ws + 3 * BD;     // [BD]   (zeroed)
  float* qq_a = ws + 4 * BD;
  float* qq_i = ws + 5 * BD;
  float* q2_a = ws + 6 * BD;
  float* q2_i = ws + 7 * BD;
  float* mu_a = ws + 8 * BD;                    // [B*NA]
  float* rs_a = mu_a + (size_t)BATCH * NA;      // [B*NA]
  float* lg_a = rs_a + (size_t)BATCH * NA;      // [B*NA]
  float* mu_i = lg_a + (size_t)BATCH * NA;      // [B*NI]
  float* rs_i = mu_i + (size_t)BATCH * NI;      // [B*NI]
  float* lg_i = rs_i + (size_t)BATCH * NI;      // [B*NI]
  unsigned* vmins = (unsigned*)(lg_i + (size_t)BATCH * NI);  // [2]

  const float scale = 0.044194173824159216f;  // 512^-0.5 folded into q2

  k_init<<<dim3((4 * BD + 255) / 256), 256, 0, stream>>>(ws, vmins);

  k_gap_stats<<<dim3(BATCH, NA / 256), 256, 0, stream>>>(audio, gap_a, mu_a, rs_a, NA);
  k_gap_stats<<<dim3(BATCH, NI / 256), 256, 0, stream>>>(image, gap_i, mu_i, rs_i, NI);

  // qq = gap @ wq^T ; q2 = qq @ wk (scale folded)
  k_gemm32<true><<<dim3(DIMD / 16, BATCH / 16), 32, 0, stream>>>(gap_a, wq, qq_a, DIMD, 1.0f);
  k_gemm32<true><<<dim3(DIMD / 16, BATCH / 16), 32, 0, stream>>>(gap_i, wq, qq_i, DIMD, 1.0f);
  k_gemm32<false><<<dim3(DIMD / 16, BATCH / 16), 32, 0, stream>>>(qq_a, wk, q2_a, DIMD, scale);
  k_gemm32<false><<<dim3(DIMD / 16, BATCH / 16), 32, 0, stream>>>(qq_i, wk, q2_i, DIMD, scale);

  k_logits<<<dim3(BATCH, NA / 256), 256, 0, stream>>>(audio, mu_a, rs_a, q2_a, lnw, lnb, lg_a, vmins + 0, NA);
  k_logits<<<dim3(BATCH, NI / 256), 256, 0, stream>>>(image, mu_i, rs_i, q2_i, lnw, lnb, lg_i, vmins + 1, NI);

  k_softmax<<<dim3(BATCH), 256, 0, stream>>>(lg_a, NA);
  k_softmax<<<dim3(BATCH), 256, 0, stream>>>(lg_i, NI);

  k_pooled<<<dim3(BATCH, NA / 256), 256, 0, stream>>>(audio, mu_a, rs_a, lg_a, lnw, lnb, vmins + 0, pool_a, NA);
  k_pooled<<<dim3(BATCH, NI / 256), 256, 0, stream>>>(image, mu_i, rs_i, lg_i, lnw, lnb, vmins + 1, pool_i, NI);

  k_finalize<<<dim3((2 * BD + 255) / 256), 256, 0, stream>>>(pool_a, 2 * BD);  // pool_a,pool_i contiguous

  // sims = pool_a @ pool_i^T -> (32,32)
  k_gemm32<true><<<dim3(2, 2), 32, 0, stream>>>(pool_a, pool_i, out, 32, 1.0f);
}